// MeshRasterizer_3985729650844
// MI455X (gfx1250) — compile-verified
//
#include <hip/hip_runtime.h>
#include <math.h>
#include <stdint.h>

#define VCOUNT 600
#define FCOUNT 1500
#define HDIM 128
#define WDIM 128
#define EPSV 1e-8f
#define NPIX (HDIM * WDIM)

#define CE_FLOATS (FCOUNT * 16)   // coefficient table (4 rows x 4 coeffs per face)
#define EV_FLOATS (FCOUNT * 8)    // edge-vertex table (x0,y0,x1,y1,x2,y2,pad,pad)
#define TOT_FLOATS (CE_FLOATS + EV_FLOATS)   // 36000 floats = 144 KB, contiguous in ws

typedef __attribute__((ext_vector_type(2))) float v2f;
typedef __attribute__((ext_vector_type(4))) float v4f;
typedef __attribute__((ext_vector_type(8))) float v8f;
typedef __attribute__((ext_vector_type(4))) unsigned int u32x4;
typedef __attribute__((ext_vector_type(8))) int i32x8;
typedef __attribute__((ext_vector_type(4))) int i32x4;

// ---------------------------------------------------------------- kernel 1
// verts_view = verts_world @ R + T ; project to NDC (x*fx/z, y*fy/z, z)
__global__ void k_transform(const float* __restrict__ vw,
                            const float* __restrict__ R,
                            const float* __restrict__ T,
                            const float* __restrict__ focal,
                            float* __restrict__ vndc) {
    int v = blockIdx.x * blockDim.x + threadIdx.x;
    if (v >= VCOUNT) return;
    float wx = vw[v * 3 + 0], wy = vw[v * 3 + 1], wz = vw[v * 3 + 2];
    float vx = wx * R[0] + wy * R[3] + wz * R[6] + T[0];
    float vy = wx * R[1] + wy * R[4] + wz * R[7] + T[1];
    float vz = wx * R[2] + wy * R[5] + wz * R[8] + T[2];
    vndc[v * 4 + 0] = focal[0] * vx / vz;
    vndc[v * 4 + 1] = focal[1] * vy / vz;
    vndc[v * 4 + 2] = vz;
    vndc[v * 4 + 3] = 0.0f;
}

// ---------------------------------------------------------------- kernel 2
// Per face: 4 affine rows (w0,w1,w2,z), each (const, px-coef, py-coef, 0),
// all pre-divided by area_safe. Degenerate faces: z-row forced to -1 so the
// face can never be valid (mirrors the |area|>EPS gate). Also store 2D verts.
__global__ void k_face_setup(const int* __restrict__ faces,
                             const float* __restrict__ vndc,
                             float* __restrict__ coef,
                             float* __restrict__ ev) {
    int f = blockIdx.x * blockDim.x + threadIdx.x;
    if (f >= FCOUNT) return;
    int i0 = faces[f * 3 + 0], i1 = faces[f * 3 + 1], i2 = faces[f * 3 + 2];
    float x0 = vndc[i0 * 4 + 0], y0 = vndc[i0 * 4 + 1], z0 = vndc[i0 * 4 + 2];
    float x1 = vndc[i1 * 4 + 0], y1 = vndc[i1 * 4 + 1], z1 = vndc[i1 * 4 + 2];
    float x2 = vndc[i2 * 4 + 0], y2 = vndc[i2 * 4 + 1], z2 = vndc[i2 * 4 + 2];

    float area = (x1 - x0) * (y2 - y0) - (y1 - y0) * (x2 - x0);
    bool ok = fabsf(area) > EPSV;
    float as = ok ? area : EPSV;

    // w0 from area2(v1, v2, p); w1 from area2(v2, v0, p); w2 from area2(v0, v1, p)
    float A0 = ((y2 - y1) * x1 - (x2 - x1) * y1) / as;
    float B0 = (-(y2 - y1)) / as;
    float C0 = (x2 - x1) / as;
    float A1 = ((y0 - y2) * x2 - (x0 - x2) * y2) / as;
    float B1 = (-(y0 - y2)) / as;
    float C1 = (x0 - x2) / as;
    float A2 = ((y1 - y0) * x0 - (x1 - x0) * y0) / as;
    float B2 = (-(y1 - y0)) / as;
    float C2 = (x1 - x0) / as;

    float Az = ok ? (A0 * z0 + A1 * z1 + A2 * z2) : -1.0f;
    float Bz = ok ? (B0 * z0 + B1 * z1 + B2 * z2) : 0.0f;
    float Cz = ok ? (C0 * z0 + C1 * z1 + C2 * z2) : 0.0f;

    float* c = coef + f * 16;
    c[0] = A0;  c[1] = B0;  c[2] = C0;  c[3] = 0.0f;
    c[4] = A1;  c[5] = B1;  c[6] = C1;  c[7] = 0.0f;
    c[8] = A2;  c[9] = B2;  c[10] = C2; c[11] = 0.0f;
    c[12] = Az; c[13] = Bz; c[14] = Cz; c[15] = 0.0f;

    float* e = ev + f * 8;
    e[0] = x0; e[1] = y0; e[2] = x1; e[3] = y1; e[4] = x2; e[5] = y2;
    e[6] = 0.0f; e[7] = 0.0f;
}

// ---------------------------------------------------------------- kernel 3
__device__ __forceinline__ float seg_d2(float ax, float ay, float bx, float by,
                                        float px, float py) {
    float dx = bx - ax, dy = by - ay;
    float l2 = fmaxf(dx * dx + dy * dy, EPSV);
    float t = ((px - ax) * dx + (py - ay) * dy) / l2;
    t = fminf(fmaxf(t, 0.0f), 1.0f);
    float qx = ax + t * dx, qy = ay + t * dy;
    float ux = px - qx, uy = py - qy;
    return ux * ux + uy * uy;
}

// One wave = 16 pixels. Lane halves split each 4-face group 2 ways.
// A fragment (16x4 f32): lane L -> row M = L%16 (face = M/4, func = M%4),
//   holds K = {2*(L/16), 2*(L/16)+1}  -> contiguous float2 in coef table (LDS).
// B fragment (4x16 f32): lane L -> col N = L%16 (pixel),
//   lanes 0-15 hold (1, px); lanes 16-31 hold (py, 0).
// D (16x16 f32): lane L, vgpr g -> row M = g + 8*(L/16), col N = L%16.
__global__ __launch_bounds__(256)
void k_rasterize(const float* __restrict__ fdata,   // coef (24000 f) ++ ev (12000 f)
                 float* __restrict__ out) {
    __shared__ float sdata[TOT_FLOATS];              // 144 KB of 320 KB WGP LDS

    // ---- stage the whole per-face table into LDS once per workgroup ----
#if __has_builtin(__builtin_amdgcn_tensor_load_to_lds) && __has_builtin(__builtin_amdgcn_s_wait_tensorcnt)
    if (threadIdx.x < 32) {                          // one TDM op per workgroup
        // flat shared addr low 32 bits == LDS byte offset (LDS aperture mapping)
        unsigned ldsOff = (unsigned)(uintptr_t)&sdata[0];
        unsigned long long ga = (unsigned long long)(uintptr_t)fdata;
        const unsigned NE = TOT_FLOATS;              // 36000 4-byte elements
        u32x4 g0;
        g0.x = 1u;                                   // count=1, user descriptor
        g0.y = ldsOff;                               // lds_addr (bytes)
        g0.z = (unsigned)(ga & 0xFFFFFFFFull);       // global_addr[31:0]
        g0.w = (unsigned)((ga >> 32) & 0x1FFFFFFull) // global_addr[56:32]
               | 0x80000000u;                        // type=2 ("image")
        i32x8 g1;
        g1[0] = (int)(2u << 16);                     // data_size=2 (4 bytes)
        g1[1] = (int)((NE & 0xFFFFu) << 16);         // tensor_dim0[15:0]
        g1[2] = (int)(((NE >> 16) & 0xFFFFu)         // tensor_dim0[31:16]
                      | (1u << 16));                 // tensor_dim1 = 1
        g1[3] = (int)((NE & 0xFFFFu) << 16);         // tile_dim0 = NE (1-D tile)
        g1[4] = 1;                                   // tile_dim1=1, tile_dim2=0
        g1[5] = (int)NE;                             // tensor_dim0_stride[31:0]
        g1[6] = 0;                                   // stride hi / dim1_stride
        g1[7] = 0;
        i32x4 gz4 = {0, 0, 0, 0};                    // groups 2/3 unused (<=2D)
        i32x8 gz8 = {0, 0, 0, 0, 0, 0, 0, 0};        // extra words (6-arg variant)
        __builtin_amdgcn_tensor_load_to_lds(g0, g1, gz4, gz4, gz8, 0);
        __builtin_amdgcn_s_wait_tensorcnt(0);
    }
    __syncthreads();
#else
    for (int i = threadIdx.x; i < TOT_FLOATS / 4; i += 256)
        ((v4f*)sdata)[i] = ((const v4f*)fdata)[i];
    __syncthreads();
#endif

    const int lane = threadIdx.x & 31;
    const int hi = lane >> 4;                 // which lane-half
    const int waveId = blockIdx.x * 8 + (threadIdx.x >> 5);
    const int p = waveId * 16 + (lane & 15);  // pixel owned by this lane
    const int pyi = p >> 7, pxi = p & 127;
    const float fpx = 1.0f - (pxi + 0.5f) * (2.0f / WDIM);
    const float fpy = 1.0f - (pyi + 0.5f) * (2.0f / HDIM);

    v2f bfrag;
    bfrag.x = hi ? fpy : 1.0f;
    bfrag.y = hi ? 0.0f : fpx;

    const int aoff = (lane & 15) * 4 + hi * 2;
    const float INF = __builtin_inff();

    float bz = INF;
    int bidx = 0;
    float bw0 = 0.0f, bw1 = 0.0f, bw2 = 0.0f;

    for (int g = 0; g < FCOUNT / 4; ++g) {
        v2f afrag = *(const v2f*)(sdata + g * 64 + aoff);   // ds_load_b64
        v8f acc = {};
        acc = __builtin_amdgcn_wmma_f32_16x16x4_f32(
            /*neg_a=*/false, afrag, /*neg_b=*/false, bfrag,
            /*c_mod=*/(short)0, acc, /*reuse_a=*/false, /*reuse_b=*/false);

        const int f0 = g * 4 + hi * 2;
        // face f0 : rows 0..3 of this half
        {
            float w0 = acc[0], w1 = acc[1], w2 = acc[2], z = acc[3];
            float m = fminf(fminf(w0, w1), w2);
            bool valid = (m >= 0.0f) && (z > 0.0f);
            float zc = valid ? z : INF;
            bool better = zc < bz;                    // strict: first min wins
            bz  = better ? zc : bz;
            bidx = better ? f0 : bidx;
            bw0 = better ? w0 : bw0;
            bw1 = better ? w1 : bw1;
            bw2 = better ? w2 : bw2;
        }
        // face f0+1 : rows 4..7 of this half
        {
            float w0 = acc[4], w1 = acc[5], w2 = acc[6], z = acc[7];
            float m = fminf(fminf(w0, w1), w2);
            bool valid = (m >= 0.0f) && (z > 0.0f);
            float zc = valid ? z : INF;
            bool better = zc < bz;
            bz  = better ? zc : bz;
            bidx = better ? (f0 + 1) : bidx;
            bw0 = better ? w0 : bw0;
            bw1 = better ? w1 : bw1;
            bw2 = better ? w2 : bw2;
        }
    }

    // merge the two lane-halves (disjoint face subsets, same pixel)
    float oz = __shfl_xor(bz, 16, 32);
    int   oi = __shfl_xor(bidx, 16, 32);
    float o0 = __shfl_xor(bw0, 16, 32);
    float o1 = __shfl_xor(bw1, 16, 32);
    float o2 = __shfl_xor(bw2, 16, 32);
    bool take = (oz < bz) || ((oz == bz) && (oi < bidx));   // argmin-first tie-break
    bz = take ? oz : bz;  bidx = take ? oi : bidx;
    bw0 = take ? o0 : bw0; bw1 = take ? o1 : bw1; bw2 = take ? o2 : bw2;

    if (hi) return;                         // lanes 0-15 own the stores

    float* pix  = out;                      // (B,H,W,1)   16384
    float* zbuf = out + NPIX;               // (B,H,W,1)   16384
    float* bary = out + 2 * NPIX;           // (B,H,W,1,3) 49152
    float* dist = out + 2 * NPIX + 3 * NPIX;// (B,H,W,1)   16384

    bool hit = bz < INF;
    if (!hit) {
        pix[p] = -1.0f; zbuf[p] = -1.0f; dist[p] = -1.0f;
        bary[p * 3 + 0] = -1.0f; bary[p * 3 + 1] = -1.0f; bary[p * 3 + 2] = -1.0f;
        return;
    }
    // winning face is always 'inside' => dist = -min edge distance^2
    const float* e = sdata + CE_FLOATS + bidx * 8;   // LDS gather
    float ax = e[0], ay = e[1], bx = e[2], by = e[3], cx = e[4], cy = e[5];
    float d0 = seg_d2(ax, ay, bx, by, fpx, fpy);
    float d1 = seg_d2(bx, by, cx, cy, fpx, fpy);
    float d2 = seg_d2(cx, cy, ax, ay, fpx, fpy);
    float md = fminf(fminf(d0, d1), d2);

    pix[p]  = (float)bidx;
    zbuf[p] = bz;
    bary[p * 3 + 0] = bw0;
    bary[p * 3 + 1] = bw1;
    bary[p * 3 + 2] = bw2;
    dist[p] = -md;
}

// ---------------------------------------------------------------- launcher
extern "C" void kernel_launch(void* const* d_in, const int* in_sizes, int n_in,
                              void* d_out, int out_size, void* d_ws, size_t ws_size,
                              hipStream_t stream) {
    (void)in_sizes; (void)n_in; (void)out_size; (void)ws_size;
    const float* vw    = (const float*)d_in[0];   // (1,600,3)
    const int*   faces = (const int*)  d_in[1];   // (1500,3)
    const float* R     = (const float*)d_in[2];   // (1,3,3)
    const float* T     = (const float*)d_in[3];   // (1,3)
    const float* focal = (const float*)d_in[4];   // (1,2)
    float* out = (float*)d_out;

    float* ws   = (float*)d_ws;
    float* vndc = ws;                       // 600*4   floats
    float* coef = vndc + VCOUNT * 4;        // 1500*16 floats
    float* ev   = coef + CE_FLOATS;         // 1500*8  floats (contiguous after coef)

    k_transform<<<(VCOUNT + 255) / 256, 256, 0, stream>>>(vw, R, T, focal, vndc);
    k_face_setup<<<(FCOUNT + 255) / 256, 256, 0, stream>>>(faces, vndc, coef, ev);
    k_rasterize<<<NPIX / (16 * 8), 256, 0, stream>>>(coef, out);
}